// GAT_38525856645644
// MI455X (gfx1250) — compile-verified
//
#include <hip/hip_runtime.h>
#include <math.h>

// ---------------------------------------------------------------------------
// GAT (2-layer, 8 heads x 16 dim) for MI455X / gfx1250.
//   - GEMMs use V_WMMA_F32_16X16X4_F32 (exact fp32 matrix path, wave32).
//   - Edge softmax/aggregation via 3 atomic passes (L2-resident).
// ---------------------------------------------------------------------------

#define N_NODES 50000
#define N_EDGES 800000
#define HEADS   8
#define HEAD_DIM 16
#define FDIM    128                 // HEADS*HEAD_DIM == C_IN
#define ETOT    (N_EDGES + N_NODES) // edges + self loops
#define NEG_SLOPE 0.2f

typedef __attribute__((ext_vector_type(2))) float v2f;
typedef __attribute__((ext_vector_type(8))) float v8f;

// ---------------- GEMM: out[N,128] = A[N,128] @ W[128,128] ------------------
// One wave computes one 16x16 output tile. K loop: 32 x wmma_f32_16x16x4_f32.
// A operand (16x4 f32): lane<16 -> row=lane, K={k0,k0+1}; lane>=16 -> row=lane-16,
// K={k0+2,k0+3}. B operand (4x16): symmetric with lanes indexing columns.
__global__ void __launch_bounds__(128)
gemm128_wmma_f32(const float* __restrict__ A, const float* __restrict__ W,
                 float* __restrict__ out) {
  const int wave = threadIdx.x >> 5;
  const int lane = threadIdx.x & 31;
  const int tile = blockIdx.x * 4 + wave; // tiles: (N/16) row-tiles x 8 col-tiles
  const int tm   = tile >> 3;
  const int tn   = tile & 7;
  const int half = lane >> 4; // 0 or 1
  const int l    = lane & 15;

  const float* __restrict__ arow = A + (size_t)(tm * 16 + l) * FDIM;
  const float* __restrict__ bcol = W + tn * 16 + l;

  v8f c = {};
#pragma unroll
  for (int k0 = 0; k0 < FDIM; k0 += 4) {
    const int ka = k0 + 2 * half;
    v2f a, b;
    a.x = arow[ka];
    a.y = arow[ka + 1];
    b.x = bcol[(size_t)ka * FDIM];
    b.y = bcol[(size_t)(ka + 1) * FDIM];
    c = __builtin_amdgcn_wmma_f32_16x16x4_f32(false, a, false, b, (short)0, c,
                                              false, false);
  }
  float* __restrict__ orow =
      out + (size_t)(tm * 16 + 8 * half) * FDIM + tn * 16 + l;
#pragma unroll
  for (int r = 0; r < 8; ++r) orow[(size_t)r * FDIM] = c[r];
}

// ---------------- per-(node,head) attention scores --------------------------
__global__ void att_scores(const float* __restrict__ h,
                           const float* __restrict__ att_src,
                           const float* __restrict__ att_dst,
                           float* __restrict__ a_src, float* __restrict__ a_dst) {
  int t = blockIdx.x * blockDim.x + threadIdx.x;
  if (t >= N_NODES * HEADS) return;
  const int hh = t & (HEADS - 1);
  const int node = t >> 3;
  const float* __restrict__ hv = h + (size_t)node * FDIM + hh * HEAD_DIM;
  const float* __restrict__ asv = att_src + hh * HEAD_DIM;
  const float* __restrict__ adv = att_dst + hh * HEAD_DIM;
  float s1 = 0.f, s2 = 0.f;
#pragma unroll
  for (int d = 0; d < HEAD_DIM; ++d) {
    float v = hv[d];
    s1 += v * asv[d];
    s2 += v * adv[d];
  }
  a_src[t] = s1;
  a_dst[t] = s2;
}

// ---------------- init kernels ---------------------------------------------
__global__ void init_ms(float* __restrict__ m, float* __restrict__ s) {
  int t = blockIdx.x * blockDim.x + threadIdx.x;
  if (t >= N_NODES * HEADS) return;
  m[t] = -__builtin_inff();
  s[t] = 0.f;
}

__global__ void init_bias(float* __restrict__ acc, const float* __restrict__ bias) {
  int t = blockIdx.x * blockDim.x + threadIdx.x;
  if (t >= N_NODES * FDIM) return;
  acc[t] = bias[t & (FDIM - 1)];
}

// ---------------- edge helpers ---------------------------------------------
__device__ __forceinline__ void edge_decode(const int* __restrict__ ei, int e,
                                            int& src, int& dst) {
  if (e < N_EDGES) {
    src = ei[e];
    dst = ei[N_EDGES + e];
  } else {
    src = dst = e - N_EDGES; // self loop
  }
}

__device__ __forceinline__ float leaky(float x) {
  return x > 0.f ? x : NEG_SLOPE * x;
}

// signed float atomic max (init must be -inf)
__device__ __forceinline__ void atomicMaxF(float* addr, float val) {
  if (val >= 0.f)
    atomicMax((int*)addr, __float_as_int(val));
  else
    atomicMin((unsigned int*)addr, __float_as_uint(val));
}

// ---------------- pass 1: segment max --------------------------------------
__global__ void edge_max(const int* __restrict__ ei,
                         const float* __restrict__ a_src,
                         const float* __restrict__ a_dst, float* __restrict__ m) {
  int t = blockIdx.x * blockDim.x + threadIdx.x;
  if (t >= ETOT * HEADS) return;
  const int hh = t & (HEADS - 1);
  const int e = t >> 3;
  int src, dst;
  edge_decode(ei, e, src, dst);
  float ee = leaky(a_src[src * HEADS + hh] + a_dst[dst * HEADS + hh]);
  atomicMaxF(&m[dst * HEADS + hh], ee);
}

// ---------------- pass 2: segment sum of exp --------------------------------
__global__ void edge_sum(const int* __restrict__ ei,
                         const float* __restrict__ a_src,
                         const float* __restrict__ a_dst,
                         const float* __restrict__ m, float* __restrict__ s) {
  int t = blockIdx.x * blockDim.x + threadIdx.x;
  if (t >= ETOT * HEADS) return;
  const int hh = t & (HEADS - 1);
  const int e = t >> 3;
  int src, dst;
  edge_decode(ei, e, src, dst);
  float ee = leaky(a_src[src * HEADS + hh] + a_dst[dst * HEADS + hh]);
  float mm = m[dst * HEADS + hh];
  if (!(mm > -__builtin_inff())) mm = 0.f; // isfinite guard
  atomicAdd(&s[dst * HEADS + hh], expf(ee - mm));
}

// ---------------- pass 3: alpha-weighted aggregation ------------------------
__global__ void edge_agg(const int* __restrict__ ei,
                         const float* __restrict__ a_src,
                         const float* __restrict__ a_dst,
                         const float* __restrict__ m, const float* __restrict__ s,
                         const float* __restrict__ h, float* __restrict__ acc) {
  int t = blockIdx.x * blockDim.x + threadIdx.x;
  if (t >= ETOT * HEADS) return;
  const int hh = t & (HEADS - 1);
  const int e = t >> 3;
  int src, dst;
  edge_decode(ei, e, src, dst);
  float ee = leaky(a_src[src * HEADS + hh] + a_dst[dst * HEADS + hh]);
  float mm = m[dst * HEADS + hh];
  if (!(mm > -__builtin_inff())) mm = 0.f;
  float p = expf(ee - mm);
  float alpha = p / (s[dst * HEADS + hh] + 1e-16f);

  const float* __restrict__ hs = h + (size_t)src * FDIM + hh * HEAD_DIM;
  float* __restrict__ od = acc + (size_t)dst * FDIM + hh * HEAD_DIM;
#pragma unroll
  for (int d = 0; d < HEAD_DIM; d += 4) {
    float4 v = *(const float4*)(hs + d);
    atomicAdd(od + d + 0, alpha * v.x);
    atomicAdd(od + d + 1, alpha * v.y);
    atomicAdd(od + d + 2, alpha * v.z);
    atomicAdd(od + d + 3, alpha * v.w);
  }
}

// ---------------- ELU -------------------------------------------------------
__global__ void elu_inplace(float* __restrict__ x) {
  int t = blockIdx.x * blockDim.x + threadIdx.x;
  if (t >= N_NODES * FDIM) return;
  float v = x[t];
  x[t] = v > 0.f ? v : (expf(v) - 1.f);
}

// ---------------------------------------------------------------------------
extern "C" void kernel_launch(void* const* d_in, const int* in_sizes, int n_in,
                              void* d_out, int out_size, void* d_ws,
                              size_t ws_size, hipStream_t stream) {
  (void)in_sizes; (void)n_in; (void)out_size;

  const float* x   = (const float*)d_in[0];
  const int*   ei  = (const int*)d_in[1]; // [2, E] flattened
  const float* W1  = (const float*)d_in[2];
  const float* as1 = (const float*)d_in[3];
  const float* ad1 = (const float*)d_in[4];
  const float* b1  = (const float*)d_in[5];
  const float* W2  = (const float*)d_in[6];
  const float* as2 = (const float*)d_in[7];
  const float* ad2 = (const float*)d_in[8];
  const float* b2  = (const float*)d_in[9];
  float* out = (float*)d_out;

  // workspace layout (floats): h[N*128] | agg[N*128] | a_src | a_dst | m | s
  float* ws    = (float*)d_ws;
  float* h     = ws;
  float* agg   = h + (size_t)N_NODES * FDIM;
  float* a_src = agg + (size_t)N_NODES * FDIM;
  float* a_dst = a_src + (size_t)N_NODES * HEADS;
  float* m     = a_dst + (size_t)N_NODES * HEADS;
  float* s     = m + (size_t)N_NODES * HEADS;
  const size_t need = ((size_t)N_NODES * FDIM * 2 + (size_t)N_NODES * HEADS * 4) *
                      sizeof(float);
  if (ws_size < need) return;

  const int gemmBlocks = (N_NODES / 16) * (FDIM / 16) / 4; // 4 waves/block
  const int nh = N_NODES * HEADS;
  const int nf = N_NODES * FDIM;
  const int eh = ETOT * HEADS;
  const int gNH = (nh + 255) / 256;
  const int gNF = (nf + 255) / 256;
  const int gEH = (eh + 255) / 256;
  dim3 B128(128), B256(256);

  // ---------------- layer 1 ----------------
  gemm128_wmma_f32<<<gemmBlocks, B128, 0, stream>>>(x, W1, h);
  att_scores<<<gNH, B256, 0, stream>>>(h, as1, ad1, a_src, a_dst);
  init_ms<<<gNH, B256, 0, stream>>>(m, s);
  init_bias<<<gNF, B256, 0, stream>>>(agg, b1);
  edge_max<<<gEH, B256, 0, stream>>>(ei, a_src, a_dst, m);
  edge_sum<<<gEH, B256, 0, stream>>>(ei, a_src, a_dst, m, s);
  edge_agg<<<gEH, B256, 0, stream>>>(ei, a_src, a_dst, m, s, h, agg);
  elu_inplace<<<gNF, B256, 0, stream>>>(agg);

  // ---------------- layer 2 ----------------
  gemm128_wmma_f32<<<gemmBlocks, B128, 0, stream>>>(agg, W2, h);
  att_scores<<<gNH, B256, 0, stream>>>(h, as2, ad2, a_src, a_dst);
  init_ms<<<gNH, B256, 0, stream>>>(m, s);
  init_bias<<<gNF, B256, 0, stream>>>(out, b2);
  edge_max<<<gEH, B256, 0, stream>>>(ei, a_src, a_dst, m);
  edge_sum<<<gEH, B256, 0, stream>>>(ei, a_src, a_dst, m, s);
  edge_agg<<<gEH, B256, 0, stream>>>(ei, a_src, a_dst, m, s, h, out);
}